// Intra_EPI_Transformer_12146167513519
// MI455X (gfx1250) — compile-verified
//
#include <hip/hip_runtime.h>
#include <math.h>

typedef __attribute__((ext_vector_type(2))) float v2f;
typedef __attribute__((ext_vector_type(8))) float v8f;

#define BN_ 4
#define C_  32
#define NN_ 4
#define VV_ 32
#define WW_ 48
#define LL_ 1536          /* VV_*WW_ */
#define HH_ 4
#define DD_ 8
#define HID_ 85
#define NT_ 96            /* LL_/16 */

static __device__ __forceinline__ float gelu_exact(float x) {
  return 0.5f * x * (1.0f + erff(x * 0.70710678118654752f));
}

// ---------------------------------------------------------------------------
// K1: permute (B,C,N,V,W)->(BN,C,L), LayerNorm over C, qkv 1x1 conv (32->96)
// ---------------------------------------------------------------------------
__global__ __launch_bounds__(256) void k1_ln_qkv(
    const float* __restrict__ buf, const float* __restrict__ n1w,
    const float* __restrict__ n1b, const float* __restrict__ qkvw,
    float* __restrict__ xperm, float* __restrict__ qkv0) {
  __shared__ float wq[96 * 32];
  for (int i = threadIdx.x; i < 96 * 32; i += 256) wq[i] = qkvw[i];
  __syncthreads();
  int gid = blockIdx.x * 256 + threadIdx.x;       // 0 .. BN_*LL_-1
  int bn = gid / LL_, l = gid % LL_;
  float x[32];
  float mu = 0.f;
#pragma unroll
  for (int c = 0; c < 32; ++c) { x[c] = buf[(c * NN_ + bn) * LL_ + l]; mu += x[c]; }
  mu *= (1.f / 32.f);
  float var = 0.f;
#pragma unroll
  for (int c = 0; c < 32; ++c) { float d = x[c] - mu; var += d * d; }
  var *= (1.f / 32.f);
  float rs = rsqrtf(var + 1e-6f);
  float xn[32];
#pragma unroll
  for (int c = 0; c < 32; ++c) {
    xperm[(bn * 32 + c) * LL_ + l] = x[c];
    xn[c] = (x[c] - mu) * rs * n1w[c] + n1b[c];
  }
  for (int o = 0; o < 96; ++o) {
    float s = 0.f;
#pragma unroll
    for (int c = 0; c < 32; ++c) s = fmaf(wq[o * 32 + c], xn[c], s);
    qkv0[(bn * 96 + o) * LL_ + l] = s;
  }
}

// ---------------------------------------------------------------------------
// K2: depthwise 3x3 (pad 1) on 96 channels + head split + l2norm(q,k) + pack
// into (bn,h,L,8) layout matching the WMMA loads in K3.
// ---------------------------------------------------------------------------
__global__ __launch_bounds__(256) void k2_dw_pack(
    const float* __restrict__ qkv0, const float* __restrict__ dww,
    float* __restrict__ q, float* __restrict__ k, float* __restrict__ v) {
  int gid = blockIdx.x * 256 + threadIdx.x;       // BN_*HH_*LL_ = 24576
  if (gid >= BN_ * HH_ * LL_) return;
  int l = gid % LL_;
  int t = gid / LL_;
  int h = t % HH_;
  int bn = t / HH_;
  int vy = l / WW_, wx = l % WW_;
  float qv[8], kv[8], vv[8];
#pragma unroll
  for (int d = 0; d < 8; ++d) {
    float s[3] = {0.f, 0.f, 0.f};
#pragma unroll
    for (int p = 0; p < 3; ++p) {
      int ch = p * 32 + h * 8 + d;
      const float* in = qkv0 + ((size_t)(bn * 96 + ch)) * LL_;
      const float* wt = dww + ch * 9;
      float acc = 0.f;
      for (int dy = 0; dy < 3; ++dy) {
        int y = vy + dy - 1;
        if (y < 0 || y >= VV_) continue;
        for (int dx = 0; dx < 3; ++dx) {
          int xw = wx + dx - 1;
          if (xw < 0 || xw >= WW_) continue;
          acc = fmaf(wt[dy * 3 + dx], in[y * WW_ + xw], acc);
        }
      }
      s[p] = acc;
    }
    qv[d] = s[0]; kv[d] = s[1]; vv[d] = s[2];
  }
  float nq = 0.f, nk = 0.f;
#pragma unroll
  for (int d = 0; d < 8; ++d) { nq += qv[d] * qv[d]; nk += kv[d] * kv[d]; }
  float iq = 1.f / fmaxf(sqrtf(nq), 1e-12f);
  float ik = 1.f / fmaxf(sqrtf(nk), 1e-12f);
  size_t base = ((size_t)((bn * HH_ + h) * LL_) + l) * 8;
#pragma unroll
  for (int d = 0; d < 8; ++d) {
    q[base + d] = qv[d] * iq;
    k[base + d] = kv[d] * ik;
    v[base + d] = vv[d];
  }
}

// ---------------------------------------------------------------------------
// K3: fused attention. Block = one 16-row tile for one bn, 4 waves = 4 heads.
// Pass 1: logits via WMMA, LANE-LOCAL online (max,sum) per row -- no cross-
//         lane ops in the loop; one 4-step shuffle merge at the end.
// Pass 2: recompute logits, softmax + relu^2/gelu head-mix modulation via
//         LDS, feed 16x16 attn tile into attn*v WMMAs.
// ---------------------------------------------------------------------------
#define WMMA_F32(A, B, Cacc) \
  __builtin_amdgcn_wmma_f32_16x16x4_f32(false, (A), false, (B), (short)0, (Cacc), false, false)

__global__ __launch_bounds__(128) void k3_attn(
    const float* __restrict__ q, const float* __restrict__ k,
    const float* __restrict__ v, const float* __restrict__ temp,
    const float* __restrict__ attca, float* __restrict__ out) {
  __shared__ float lds_a1[HH_ * 256];
  __shared__ float lds_at[HH_ * 256];
  const int h = threadIdx.x >> 5;        // wave == head
  const int lane = threadIdx.x & 31;
  const int cc = lane & 15;
  const int hi = lane >> 4;
  const int bn = blockIdx.x / NT_;
  const int it = blockIdx.x % NT_;
  const int i0 = it * 16;
  const float* qh = q + ((size_t)((bn * HH_ + h) * LL_)) * 8;
  const float* kh = k + ((size_t)((bn * HH_ + h) * LL_)) * 8;
  const float* vh = v + ((size_t)((bn * HH_ + h) * LL_)) * 8;
  const float tp = temp[h];
  float wsc[HH_], wsh[HH_];
#pragma unroll
  for (int j = 0; j < HH_; ++j) {
    wsc[j] = attca[h * HH_ + j];
    wsh[j] = attca[(HH_ + h) * HH_ + j];
  }
  // A-matrix (q rows): lane m=cc holds K = chunk + 2*hi, +1  (ISA 7.12.2)
  v2f qA0 = *(const v2f*)(qh + (i0 + cc) * 8 + 2 * hi);
  v2f qA1 = *(const v2f*)(qh + (i0 + cc) * 8 + 4 + 2 * hi);

  // -------- pass 1: lane-local online max/sum (no cross-lane traffic) -----
  float lmax[8], lsum[8];
#pragma unroll
  for (int r = 0; r < 8; ++r) { lmax[r] = -3.0e38f; lsum[r] = 0.f; }
  for (int jt = 0; jt < NT_; ++jt) {
    const float* kb = kh + (jt * 16 + cc) * 8 + 2 * hi;
    v2f kB0 = *(const v2f*)(kb);
    v2f kB1 = *(const v2f*)(kb + 4);
    v8f acc = {};
    acc = WMMA_F32(qA0, kB0, acc);
    acc = WMMA_F32(qA1, kB1, acc);
#pragma unroll
    for (int r = 0; r < 8; ++r) {
      float val = acc[r] * tp;                   // row m=r+8*hi, col n=cc
      float d = val - lmax[r];
      float e = __expf(-fabsf(d));               // exp(-|d|): the non-1 factor
      lsum[r] = (d > 0.f) ? fmaf(lsum[r], e, 1.f) : (lsum[r] + e);
      lmax[r] = fmaxf(lmax[r], val);
    }
  }
  // single cross-lane (max,sum) merge within each 16-lane half
  float rmax[8], rinv[8];
#pragma unroll
  for (int r = 0; r < 8; ++r) {
    float m = lmax[r], s = lsum[r];
#pragma unroll
    for (int msk = 1; msk <= 8; msk <<= 1) {
      float mo = __shfl_xor(m, msk, 32);
      float so = __shfl_xor(s, msk, 32);
      float e = __expf(-fabsf(m - mo));
      s = (m >= mo) ? fmaf(so, e, s) : fmaf(s, e, so);
      m = fmaxf(m, mo);
    }
    rmax[r] = m;
    rinv[r] = 1.f / s;
  }

  // -------- pass 2: full attention weights + attn*v -----------------------
  v8f oacc = {};
  for (int jt = 0; jt < NT_; ++jt) {
    const int j0 = jt * 16;
    const float* kb = kh + (j0 + cc) * 8 + 2 * hi;
    v2f kB0 = *(const v2f*)(kb);
    v2f kB1 = *(const v2f*)(kb + 4);
    v8f acc = {};
    acc = WMMA_F32(qA0, kB0, acc);
    acc = WMMA_F32(qA1, kB1, acc);
    float lg[8];
#pragma unroll
    for (int r = 0; r < 8; ++r) {
      lg[r] = acc[r] * tp;
      float x = fmaxf(lg[r], 0.f);
      x = x * x;                                 // relu^2
      float a1 = gelu_exact(x) * x;              // gelu(attn1)*attn1
      lds_a1[h * 256 + (r + 8 * hi) * 16 + cc] = a1;
    }
    __syncthreads();
#pragma unroll
    for (int r = 0; r < 8; ++r) {
      int m = r + 8 * hi;
      float sc = 0.f, sh = 0.f;
#pragma unroll
      for (int j = 0; j < HH_; ++j) {            // head-mix via att_ca_w
        float a = lds_a1[j * 256 + m * 16 + cc];
        sc = fmaf(wsc[j], a, sc);
        sh = fmaf(wsh[j], a, sh);
      }
      float a0 = __expf(lg[r] - rmax[r]) * rinv[r];
      lds_at[h * 256 + m * 16 + cc] = a0 * (1.f + sc) + sh;
    }
    __syncthreads();
#pragma unroll
    for (int t = 0; t < 4; ++t) {                // attn(16x16) x v(16x8)
      int kk = 4 * t + 2 * hi;
      v2f aA;
      aA.x = lds_at[h * 256 + cc * 16 + kk];     // A: M=cc, K=kk,kk+1
      aA.y = lds_at[h * 256 + cc * 16 + kk + 1];
      v2f vB;
      vB.x = (cc < 8) ? vh[(j0 + kk) * 8 + cc] : 0.f;       // B: K=kk, N=cc
      vB.y = (cc < 8) ? vh[(j0 + kk + 1) * 8 + cc] : 0.f;
      oacc = WMMA_F32(aA, vB, oacc);
    }
  }
  // oacc[r] = out[row i0+r+8*hi][d=cc]; store as channel h*8+d spatial map
  if (cc < 8) {
#pragma unroll
    for (int r = 0; r < 8; ++r)
      out[(bn * 32 + (h * 8 + cc)) * LL_ + (i0 + r + 8 * hi)] = oacc[r];
  }
}

// ---------------------------------------------------------------------------
// K4: proj 1x1 (32->32) + residual + LayerNorm2 + ffn_in 1x1 (32->170)
// ---------------------------------------------------------------------------
__global__ __launch_bounds__(256) void k4_proj_ffnin(
    const float* __restrict__ ao, const float* __restrict__ xperm,
    const float* __restrict__ projw, const float* __restrict__ n2w,
    const float* __restrict__ n2b, const float* __restrict__ ffiw,
    float* __restrict__ x2, float* __restrict__ h0) {
  __shared__ float wp[32 * 32];
  __shared__ float wi[170 * 32];
  for (int i = threadIdx.x; i < 32 * 32; i += 256) wp[i] = projw[i];
  for (int i = threadIdx.x; i < 170 * 32; i += 256) wi[i] = ffiw[i];
  __syncthreads();
  int gid = blockIdx.x * 256 + threadIdx.x;
  int bn = gid / LL_, l = gid % LL_;
  float a[32];
#pragma unroll
  for (int c = 0; c < 32; ++c) a[c] = ao[(bn * 32 + c) * LL_ + l];
  float y[32];
  float mu = 0.f;
#pragma unroll
  for (int c = 0; c < 32; ++c) {
    float s = 0.f;
#pragma unroll
    for (int c2 = 0; c2 < 32; ++c2) s = fmaf(wp[c * 32 + c2], a[c2], s);
    y[c] = xperm[(bn * 32 + c) * LL_ + l] + s;
    x2[(bn * 32 + c) * LL_ + l] = y[c];
    mu += y[c];
  }
  mu *= (1.f / 32.f);
  float var = 0.f;
#pragma unroll
  for (int c = 0; c < 32; ++c) { float d = y[c] - mu; var += d * d; }
  var *= (1.f / 32.f);
  float rs = rsqrtf(var + 1e-6f);
  float z[32];
#pragma unroll
  for (int c = 0; c < 32; ++c) z[c] = (y[c] - mu) * rs * n2w[c] + n2b[c];
  for (int o = 0; o < 170; ++o) {
    float s = 0.f;
#pragma unroll
    for (int c = 0; c < 32; ++c) s = fmaf(wi[o * 32 + c], z[c], s);
    h0[((size_t)(bn * 170 + o)) * LL_ + l] = s;
  }
}

// ---------------------------------------------------------------------------
// K5: depthwise 3x3 (pad 1) on 170 channels + gelu gate (gelu(h1)*h2)
// ---------------------------------------------------------------------------
__global__ __launch_bounds__(256) void k5_ffndw(
    const float* __restrict__ h0, const float* __restrict__ dww,
    float* __restrict__ hg) {
  int gid = blockIdx.x * 256 + threadIdx.x;       // BN_*85*LL_
  if (gid >= BN_ * HID_ * LL_) return;
  int l = gid % LL_;
  int t = gid / LL_;
  int c = t % HID_;
  int bn = t / HID_;
  int vy = l / WW_, wx = l % WW_;
  float r[2];
#pragma unroll
  for (int p = 0; p < 2; ++p) {
    int ch = p * HID_ + c;
    const float* in = h0 + ((size_t)(bn * 170 + ch)) * LL_;
    const float* wt = dww + ch * 9;
    float acc = 0.f;
    for (int dy = 0; dy < 3; ++dy) {
      int y = vy + dy - 1;
      if (y < 0 || y >= VV_) continue;
      for (int dx = 0; dx < 3; ++dx) {
        int xw = wx + dx - 1;
        if (xw < 0 || xw >= WW_) continue;
        acc = fmaf(wt[dy * 3 + dx], in[y * WW_ + xw], acc);
      }
    }
    r[p] = acc;
  }
  hg[((size_t)(bn * HID_ + c)) * LL_ + l] = gelu_exact(r[0]) * r[1];
}

// ---------------------------------------------------------------------------
// K6: ffn_out 1x1 (85->32) + residual + final permute to (B,C,N,V,W)
// ---------------------------------------------------------------------------
__global__ __launch_bounds__(256) void k6_ffnout(
    const float* __restrict__ hg, const float* __restrict__ x2,
    const float* __restrict__ fow, float* __restrict__ out) {
  __shared__ float wo[32 * HID_];
  for (int i = threadIdx.x; i < 32 * HID_; i += 256) wo[i] = fow[i];
  __syncthreads();
  int gid = blockIdx.x * 256 + threadIdx.x;
  int bn = gid / LL_, l = gid % LL_;
  float g[HID_];
  for (int c2 = 0; c2 < HID_; ++c2) g[c2] = hg[((size_t)(bn * HID_ + c2)) * LL_ + l];
#pragma unroll
  for (int c = 0; c < 32; ++c) {
    float s = 0.f;
    for (int c2 = 0; c2 < HID_; ++c2) s = fmaf(wo[c * HID_ + c2], g[c2], s);
    out[(c * NN_ + bn) * LL_ + l] = x2[(bn * 32 + c) * LL_ + l] + s;
  }
}

// ---------------------------------------------------------------------------
extern "C" void kernel_launch(void* const* d_in, const int* in_sizes, int n_in,
                              void* d_out, int out_size, void* d_ws, size_t ws_size,
                              hipStream_t stream) {
  (void)in_sizes; (void)n_in; (void)out_size; (void)ws_size;
  const float* buf   = (const float*)d_in[0];
  const float* temp  = (const float*)d_in[1];
  const float* n1w   = (const float*)d_in[2];
  const float* n1b   = (const float*)d_in[3];
  const float* qkvw  = (const float*)d_in[4];
  const float* dww   = (const float*)d_in[5];
  const float* projw = (const float*)d_in[6];
  const float* attca = (const float*)d_in[7];
  const float* n2w   = (const float*)d_in[8];
  const float* n2b   = (const float*)d_in[9];
  const float* ffiw  = (const float*)d_in[10];
  const float* ffdw  = (const float*)d_in[11];
  const float* ffow  = (const float*)d_in[12];

  float* ws = (float*)d_ws;
  float* xperm = ws;                 // 196608  (BN,C,L)     live: K1..K4
  float* qkv0  = ws + 196608;        // 589824  (BN,96,L)    live: K1..K2
  float* qb    = ws + 786432;        // 196608  (BN,H,L,8)   live: K2..K3
  float* kb    = ws + 983040;        // 196608
  float* vb    = ws + 1179648;       // 196608
  float* aout  = ws + 1376256;       // 196608  attn out     live: K3..K4
  float* x2    = ws + 1572864;       // 196608  residual     live: K4..K6
  float* h0    = ws + 196608;        // 1044480 (reuses qkv0+qkv, dead after K3)
  float* hg    = ws + 1769472;       // 522240               live: K5..K6
  // total workspace: 2291712 floats ~= 8.8 MB

  k1_ln_qkv<<<(BN_ * LL_) / 256, 256, 0, stream>>>(buf, n1w, n1b, qkvw, xperm, qkv0);
  k2_dw_pack<<<(BN_ * HH_ * LL_ + 255) / 256, 256, 0, stream>>>(qkv0, dww, qb, kb, vb);
  k3_attn<<<BN_ * NT_, 128, 0, stream>>>(qb, kb, vb, temp, attca, aout);
  k4_proj_ffnin<<<(BN_ * LL_) / 256, 256, 0, stream>>>(aout, xperm, projw, n2w, n2b, ffiw, x2, h0);
  k5_ffndw<<<(BN_ * HID_ * LL_ + 255) / 256, 256, 0, stream>>>(h0, ffdw, hg);
  k6_ffnout<<<(BN_ * LL_) / 256, 256, 0, stream>>>(hg, x2, ffow, (float*)d_out);
}